// ViscRNNCellGen_69114613727510
// MI455X (gfx1250) — compile-verified
//
#include <hip/hip_runtime.h>
#include <hip/hip_bf16.h>

// Shapes fixed by the reference: x:(64,4096,2) f32, tau:(1,256) f32,
// out:(64,4096,257) f32.
#define B_LEN 64
#define T_LEN 4096
#define U_LEN 256
#define OUT_C (U_LEN + 1)

__device__ __forceinline__ float fast_exp2(float x) {
#if __has_builtin(__builtin_amdgcn_exp2f)
  return __builtin_amdgcn_exp2f(x);     // v_exp_f32
#else
  return exp2f(x);
#endif
}

__global__ __launch_bounds__(U_LEN) void visc_rnn_scan(
    const float* __restrict__ x,    // (B, T, 2) interleaved {dt, sig0}
    const float* __restrict__ tau,  // (1, U)
    float* __restrict__ out)        // (B, T, 1+U)
{
  // 32 KB stage of x[b,:,:] for this batch (320 KB/WGP available on CDNA5).
  __shared__ float xs[2 * T_LEN];

  const int b = blockIdx.x;
  const int u = threadIdx.x;
  const float* xb = x + (size_t)b * (2 * T_LEN);

  // ---- Stage x[b] global -> LDS via CDNA5 async copy (ASYNCcnt path) ----
#if __has_builtin(__builtin_amdgcn_global_load_async_to_lds_b128)
  {
    // Builtin prototype (from hipcc diagnostic): params are
    // 'int __attribute__((vector_size(16)))*' in AS1 (global) / AS3 (LDS).
    typedef int v4i __attribute__((vector_size(16)));
    typedef __attribute__((address_space(1))) v4i GV4;  // global
    typedef __attribute__((address_space(3))) v4i LV4;  // LDS
    GV4* g = (GV4*)xb;
    LV4* l = (LV4*)xs;
    // 8192 floats = 2048 x b128 granules; 256 lanes x 8 issues, 16B per lane.
#pragma unroll
    for (int k = 0; k < (2 * T_LEN) / (U_LEN * 4); ++k) {
      const int q = u + k * U_LEN;  // 16-byte granule index
      __builtin_amdgcn_global_load_async_to_lds_b128(g + q, l + q, 0, 0);
    }
  }
#if __has_builtin(__builtin_amdgcn_s_wait_asynccnt)
  __builtin_amdgcn_s_wait_asynccnt(0);
#else
  asm volatile("s_wait_asynccnt 0" ::: "memory");
#endif
#else
  // Fallback: plain vectorized copy through VGPRs.
  for (int k = u; k < (2 * T_LEN) / 4; k += U_LEN)
    ((float4*)xs)[k] = ((const float4*)xb)[k];
#endif
  __syncthreads();

  // Per-unit decay constant: tauPos = relu(tau)+1e-5 ;
  // a = exp(-dt/(tauPos*1000)) = 2^(dt*c2), b = 2^(0.5*dt*c2) = exp(-dt/(2*tauPos*1000)).
  const float tv = fmaxf(tau[u], 0.0f) + 1e-5f;
  const float c2 = -1.4426950408889634f / (tv * 1000.0f);

  float h = 0.0f;
  float sig_prev = 0.0f;
  float* outb = out + (size_t)b * T_LEN * OUT_C;

#pragma unroll 4
  for (int t = 0; t < T_LEN; ++t) {
    const float2 v = ((const float2*)xs)[t];  // single ds_load_b64, bank-broadcast
    const float dt  = v.x;
    const float sig = v.y;
    const float dsig = sig - sig_prev;
    sig_prev = sig;

    const float e  = dt * c2;
    const float a  = fast_exp2(e);          // v_exp_f32
    const float bb = fast_exp2(0.5f * e);   // v_exp_f32
    h = fmaf(a, h, bb * dsig);

    float* orow = outb + (size_t)t * OUT_C;
    // Streaming, write-once output (257 MB > L2): non-temporal stores.
    __builtin_nontemporal_store(h, orow + 1 + u);   // coalesced across the block
    if (u == 0) __builtin_nontemporal_store(sig, orow);  // channel 0 = sig0
  }
}

extern "C" void kernel_launch(void* const* d_in, const int* in_sizes, int n_in,
                              void* d_out, int out_size, void* d_ws, size_t ws_size,
                              hipStream_t stream) {
  const float* x   = (const float*)d_in[0];  // (64, 4096, 2) f32
  const float* tau = (const float*)d_in[1];  // (1, 256) f32
  float* out = (float*)d_out;                // (64, 4096, 257) f32
  (void)in_sizes; (void)n_in; (void)out_size; (void)d_ws; (void)ws_size;

  visc_rnn_scan<<<B_LEN, U_LEN, 0, stream>>>(x, tau, out);
}